// LocalAggregator_40432822124943
// MI455X (gfx1250) — compile-verified
//
#include <hip/hip_runtime.h>

typedef float v2f __attribute__((ext_vector_type(2)));
typedef float v8f __attribute__((ext_vector_type(8)));

#define GRID_SZ   0.4f
#define SCALEMUL  3.0f
#define PCX      -40.0f
#define PCY      -40.0f
#define PCZ      -1.0f
#define EPS_SQ    1e-8f
#define CCH       17
#define PKSTRIDE  20
#define NCX       25      // 200/8
#define NCY       25
#define NCZ       4       // 16/4
#define NCELL     (NCX*NCY*NCZ)
#define GMAX      2048

__device__ __forceinline__ int cell_of(float px, float py, float pz) {
    int cx = ((int)floorf((px - PCX) / GRID_SZ)) >> 3;
    int cy = ((int)floorf((py - PCY) / GRID_SZ)) >> 3;
    int cz = ((int)floorf((pz - PCZ) / GRID_SZ)) >> 2;
    cx = min(max(cx, 0), NCX - 1);
    cy = min(max(cy, 0), NCY - 1);
    cz = min(max(cz, 0), NCZ - 1);
    return (cz * NCY + cy) * NCX + cx;
}

// ---------------- binning: histogram -> scan -> scatter permutation ----------------
__global__ __launch_bounds__(256) void la_zero(int* __restrict__ hist) {
    int i = blockIdx.x * blockDim.x + threadIdx.x;
    if (i < NCELL) hist[i] = 0;
}

__global__ __launch_bounds__(256) void la_hist(const float* __restrict__ pts,
                                               int* __restrict__ hist, int P) {
    int p = blockIdx.x * blockDim.x + threadIdx.x;
    if (p >= P) return;
    atomicAdd(&hist[cell_of(pts[p*3+0], pts[p*3+1], pts[p*3+2])], 1);
}

__global__ __launch_bounds__(256) void la_scan(const int* __restrict__ hist,
                                               int* __restrict__ ofs) {
    __shared__ int part[256];
    const int t = threadIdx.x;
    const int CHUNK = (NCELL + 255) / 256;
    const int lo = t * CHUNK, hi = min(lo + CHUNK, NCELL);
    int s = 0;
    for (int i = lo; i < hi; ++i) s += hist[i];
    part[t] = s;
    __syncthreads();
    for (int d = 1; d < 256; d <<= 1) {
        int v = (t >= d) ? part[t - d] : 0;
        __syncthreads();
        part[t] += v;
        __syncthreads();
    }
    int run = (t == 0) ? 0 : part[t - 1];
    for (int i = lo; i < hi; ++i) { ofs[i] = run; run += hist[i]; }
}

__global__ __launch_bounds__(256) void la_scatter(const float* __restrict__ pts,
                                                  int* __restrict__ ofs,
                                                  int* __restrict__ perm, int P) {
    int p = blockIdx.x * blockDim.x + threadIdx.x;
    if (p >= P) return;
    int slot = atomicAdd(&ofs[cell_of(pts[p*3+0], pts[p*3+1], pts[p*3+2])], 1);
    perm[slot] = p;
}

// ---------------- prep: bake per-gaussian params ----------------
__global__ __launch_bounds__(256) void la_prep(
    const float* __restrict__ means3D, const float* __restrict__ opas,
    const float* __restrict__ u, const float* __restrict__ v,
    const float* __restrict__ scales, const float* __restrict__ rot3D,
    float* __restrict__ pk, int4* __restrict__ meta, int G)
{
    int g = blockIdx.x * blockDim.x + threadIdx.x;
    if (g >= G) return;
    const float* R = rot3D + (size_t)g * 9;          // R[j][i], row-major
    float mx = means3D[g*3+0], my = means3D[g*3+1], mz = means3D[g*3+2];
    float* q = pk + (size_t)g * PKSTRIDE;
    float c0x=R[0], c0y=R[3], c0z=R[6];
    float c1x=R[1], c1y=R[4], c1z=R[7];
    float c2x=R[2], c2y=R[5], c2z=R[8];
    q[0]=c0x; q[1]=c0y; q[2]=c0z;
    q[3]=c1x; q[4]=c1y; q[5]=c1z;
    q[6]=c2x; q[7]=c2y; q[8]=c2z;
    q[9]  = c0x*mx + c0y*my + c0z*mz;
    q[10] = c1x*mx + c1y*my + c1z*mz;
    q[11] = c2x*mx + c2y*my + c2z*mz;
    float sx = scales[g*3+0], sy = scales[g*3+1], sz = scales[g*3+2];
    q[12] = 1.0f/sx; q[13] = 1.0f/sy; q[14] = 1.0f/sz;
    float uu = u[g], vv = v[g];
    q[15] = 1.0f/vv;      // inv_e2
    q[16] = vv/uu;        // ratio
    q[17] = 1.0f/uu;      // inv_e1
    q[18] = opas[g];
    q[19] = 0.0f;
    float smax = fmaxf(sx, fmaxf(sy, sz));
    int radius = (int)ceilf(smax * SCALEMUL / GRID_SZ);
    if (radius < 1) radius = 1;
    int4 m;
    m.x = (int)floorf((mx - PCX) / GRID_SZ);
    m.y = (int)floorf((my - PCY) / GRID_SZ);
    m.z = (int)floorf((mz - PCZ) / GRID_SZ);
    m.w = radius;
    meta[g] = m;
}

// ---------------- main: 16 clustered points per wave, K=4 gaussians per WMMA ----------------
__global__ __launch_bounds__(256) void la_main(
    const float* __restrict__ pts, const float* __restrict__ sem,
    const float* __restrict__ pk,  const int4* __restrict__ meta,
    const int* __restrict__ perm,
    float* __restrict__ out_logits, float* __restrict__ out_bin,
    float* __restrict__ out_dens, int P, int G)
{
    __shared__ int4 smeta[GMAX];
    for (int i = threadIdx.x; i < G; i += blockDim.x) smeta[i] = meta[i];
    __syncthreads();

    const int lane  = threadIdx.x & 31;
    const int wave  = threadIdx.x >> 5;
    const int pbase = blockIdx.x * 128 + wave * 16;
    const int myp   = perm[pbase + (lane & 15)];   // spatially clustered tile of 16 points

    const float px = pts[myp*3+0], py = pts[myp*3+1], pz = pts[myp*3+2];
    const int pix = (int)floorf((px - PCX) / GRID_SZ);
    const int piy = (int)floorf((py - PCY) / GRID_SZ);
    const int piz = (int)floorf((pz - PCZ) / GRID_SZ);

    v8f acc = {0.f,0.f,0.f,0.f,0.f,0.f,0.f,0.f};
    float dens = 0.f, lgs = 0.f, c16 = 0.f;
    const int koff = (lane >> 4) << 1;   // lanes 0-15 -> K0/K1, lanes 16-31 -> K2/K3
    const int c    = lane & 15;          // N (channel) for B and D

    for (int gb = 0; gb < G; gb += 4) {
        const int g0 = gb + koff;
        const int g1 = g0 + 1;
        const int4 m0 = smeta[g0];
        const int4 m1 = smeta[g1];
        const bool in0 = (max(abs(pix-m0.x), max(abs(piy-m0.y), abs(piz-m0.z))) <= m0.w);
        const bool in1 = (max(abs(pix-m1.x), max(abs(piy-m1.y), abs(piz-m1.z))) <= m1.w);
        if (!__any(in0 || in1)) continue;   // wave-uniform skip

        float a0 = 0.f, a1 = 0.f;
        if (in0) {
            const float* q = pk + (size_t)g0 * PKSTRIDE;
            float lx = q[0]*px + q[1]*py + q[2]*pz - q[9];
            float ly = q[3]*px + q[4]*py + q[5]*pz - q[10];
            float lz = q[6]*px + q[7]*py + q[8]*pz - q[11];
            float tx = lx*q[12], ty = ly*q[13], tz = lz*q[14];
            float s2x = tx*tx + EPS_SQ, s2y = ty*ty + EPS_SQ, s2z = tz*tz + EPS_SQ;
            float pe = __powf(s2x, q[15]) + __powf(s2y, q[15]);
            float f  = __powf(pe, q[16]) + __powf(s2z, q[17]);
            a0 = q[18] * __expf(-0.5f * f);
        }
        if (in1) {
            const float* q = pk + (size_t)g1 * PKSTRIDE;
            float lx = q[0]*px + q[1]*py + q[2]*pz - q[9];
            float ly = q[3]*px + q[4]*py + q[5]*pz - q[10];
            float lz = q[6]*px + q[7]*py + q[8]*pz - q[11];
            float tx = lx*q[12], ty = ly*q[13], tz = lz*q[14];
            float s2x = tx*tx + EPS_SQ, s2y = ty*ty + EPS_SQ, s2z = tz*tz + EPS_SQ;
            float pe = __powf(s2x, q[15]) + __powf(s2y, q[15]);
            float f  = __powf(pe, q[16]) + __powf(s2z, q[17]);
            a1 = q[18] * __expf(-0.5f * f);
        }

        dens += a0 + a1;
        lgs  += log1pf(-fminf(a0, 0.9999f)) + log1pf(-fminf(a1, 0.9999f));
        const float* s0 = sem + (size_t)g0 * CCH;
        const float* s1 = sem + (size_t)g1 * CCH;
        c16 += a0 * s0[16] + a1 * s1[16];

        v2f A; A.x = a0;    A.y = a1;      // 16x4 f32 A tile (2 VGPRs)
        v2f B; B.x = s0[c]; B.y = s1[c];   // 4x16 f32 B tile (semantics rows)
        acc = __builtin_amdgcn_wmma_f32_16x16x4_f32(
            false, A, false, B, (short)0, acc, false, false);
    }

    // combine the two K-halves of each point (lane L <-> lane L+16)
    const float dAll   = dens + __shfl_xor(dens, 16, 32);
    const float lgsAll = lgs  + __shfl_xor(lgs, 16, 32);
    const float c16All = c16  + __shfl_xor(c16, 16, 32);

#pragma unroll
    for (int r = 0; r < 8; ++r) {
        const int M = (lane < 16) ? r : (r + 8);          // C/D layout: vgpr r = rows r / r+8
        const float dR = __shfl(dAll, M, 32);
        const int pM = __shfl(myp, M, 32);                // original point index of row M
        out_logits[(size_t)pM * CCH + c] = acc[r] / (dR + 1e-9f);
    }
    if (lane < 16) {
        out_logits[(size_t)myp * CCH + 16] = c16All / (dAll + 1e-9f);
        out_bin[myp]  = 1.0f - __expf(lgsAll);
        out_dens[myp] = dAll;
    }
}

extern "C" void kernel_launch(void* const* d_in, const int* in_sizes, int n_in,
                              void* d_out, int out_size, void* d_ws, size_t ws_size,
                              hipStream_t stream) {
    const float* pts     = (const float*)d_in[0];
    const float* means3D = (const float*)d_in[1];
    const float* opas    = (const float*)d_in[2];
    const float* u       = (const float*)d_in[3];
    const float* v       = (const float*)d_in[4];
    const float* sem     = (const float*)d_in[5];
    const float* scales  = (const float*)d_in[6];
    const float* rot3D   = (const float*)d_in[7];
    const int P = in_sizes[0] / 3;
    const int G = in_sizes[1] / 3;

    char* ws = (char*)d_ws;
    float* pk   = (float*)ws;                       ws += (size_t)G * PKSTRIDE * sizeof(float);
    int4*  meta = (int4*)ws;                        ws += (size_t)G * sizeof(int4);
    int*   hist = (int*)ws;                         ws += (size_t)NCELL * sizeof(int);
    int*   cofs = (int*)ws;                         ws += (size_t)NCELL * sizeof(int);
    int*   perm = (int*)ws;                         ws += (size_t)P * sizeof(int);

    float* out_logits = (float*)d_out;
    float* out_bin    = out_logits + (size_t)P * CCH;
    float* out_dens   = out_bin + P;

    // spatial binning of points (rebuilt from scratch every call)
    la_zero<<<(NCELL + 255) / 256, 256, 0, stream>>>(hist);
    la_hist<<<(P + 255) / 256, 256, 0, stream>>>(pts, hist, P);
    la_scan<<<1, 256, 0, stream>>>(hist, cofs);
    la_scatter<<<(P + 255) / 256, 256, 0, stream>>>(pts, cofs, perm, P);

    la_prep<<<(G + 255) / 256, 256, 0, stream>>>(means3D, opas, u, v, scales, rot3D,
                                                 pk, meta, G);
    la_main<<<P / 128, 256, 0, stream>>>(pts, sem, pk, meta, perm,
                                         out_logits, out_bin, out_dens, P, G);
}